// LLaDA2MoeDecoderLayer_89446988907019
// MI455X (gfx1250) — compile-verified
//
#include <hip/hip_runtime.h>
#include <cstdint>

// ---------------------------------------------------------------------------
// Shapes (compile-time constants from the reference)
// ---------------------------------------------------------------------------
constexpr int Tn   = 1024;   // tokens (B*S)
constexpr int Hn   = 2048;   // hidden
constexpr int NH   = 16;     // attention heads
constexpr int HD   = 128;    // head dim
constexpr int NKV  = 4;      // kv heads
constexpr int QKVD = (NH + 2 * NKV) * HD;   // 3072
constexpr int ROT  = 64;
constexpr int En   = 16;     // experts
constexpr int Kt   = 4;      // top-k
constexpr int MIn  = 1024;   // expert intermediate
constexpr int NPAIR = Tn * Kt;              // 4096 token-expert pairs
constexpr float EPS = 1e-6f;

typedef float v2f __attribute__((ext_vector_type(2)));
typedef float v8f __attribute__((ext_vector_type(8)));

// ---------------------------------------------------------------------------
// CDNA5 async global->LDS copies (ASYNCcnt-tracked, bypass VGPRs).
// LDS operand is the wave-relative LDS byte address == low 32 bits of the
// generic pointer (flat->LDS truncation per ISA 10.2).
// ---------------------------------------------------------------------------
__device__ __forceinline__ void async_copy_b128(const float* g, float* l) {
    asm volatile("global_load_async_to_lds_b128 %0, %1, off"
                 :: "v"((uint32_t)(uintptr_t)l), "v"(g) : "memory");
}
__device__ __forceinline__ void async_copy_b64(const float* g, float* l) {
    asm volatile("global_load_async_to_lds_b64 %0, %1, off"
                 :: "v"((uint32_t)(uintptr_t)l), "v"(g) : "memory");
}
__device__ __forceinline__ void wait_async0() {
    asm volatile("s_wait_asynccnt 0" ::: "memory");
}

// ---------------------------------------------------------------------------
// Generic WMMA GEMM:  C[M x N] = A[M x K] * B[N x K]^T  (+ optional addends)
// block = 256 threads (8 waves); tile M=16, N=128; K double-buffered in LDS
// via async global->LDS DMA, overlapped with the f32 WMMA pipe.
// ---------------------------------------------------------------------------
enum { GM_STORE = 0, GM_ADD = 1, GM_FINAL = 2 };

template <int MODE>
__global__ __launch_bounds__(256) void gemm16_kernel(
    const float* __restrict__ A, const float* __restrict__ Bm,
    float* __restrict__ C, int M, int N, int Kd,
    const float* __restrict__ add0, const float* __restrict__ add1)
{
    __shared__ __align__(16) float As[2][16][36];
    __shared__ __align__(16) float Bs[2][128][36];
    const int tid  = threadIdx.x;
    const int wave = tid >> 5, lane = tid & 31;
    const int half = lane >> 4, idx = lane & 15;
    const int m0 = blockIdx.y * 16;
    const int n0 = blockIdx.x * 128;

    // fixed per-thread staging assignment
    const int ar = (tid << 1) >> 5, ac = (tid << 1) & 31;     // A: b64 per thread
    const float* arow = A + (size_t)(m0 + ar) * Kd + ac;

    auto issue = [&](int buf, int k0) {
        async_copy_b64(arow + k0, &As[buf][ar][ac]);
#pragma unroll
        for (int i = 0; i < 4; i++) {                          // B: 4 x b128
            int f = tid + i * 256;
            int r = f >> 3, c4 = (f & 7) << 2;
            async_copy_b128(Bm + (size_t)(n0 + r) * Kd + k0 + c4, &Bs[buf][r][c4]);
        }
    };

    v8f acc = {};
    const int nch = Kd >> 5;
    issue(0, 0);
    for (int ch = 0; ch < nch; ch++) {
        wait_async0();
        __syncthreads();
        if (ch + 1 < nch) issue((ch + 1) & 1, (ch + 1) << 5);
        const int buf = ch & 1;
#pragma unroll
        for (int kk = 0; kk < 32; kk += 4) {
            v2f a, b;
            a[0] = As[buf][idx][kk + 2 * half];
            a[1] = As[buf][idx][kk + 2 * half + 1];
            b[0] = Bs[buf][wave * 16 + idx][kk + 2 * half];
            b[1] = Bs[buf][wave * 16 + idx][kk + 2 * half + 1];
            acc = __builtin_amdgcn_wmma_f32_16x16x4_f32(
                false, a, false, b, (short)0, acc, false, false);
        }
    }
#pragma unroll
    for (int r = 0; r < 8; r++) {
        int m = m0 + r + 8 * half;
        int n = n0 + wave * 16 + idx;
        size_t off = (size_t)m * N + n;
        float v = acc[r];
        if (MODE == GM_ADD)   v += add0[off];
        if (MODE == GM_FINAL) v += add0[off] + add1[off];
        C[off] = v;
    }
}

// ---------------------------------------------------------------------------
// RMSNorm over rows of length Hc
// ---------------------------------------------------------------------------
__global__ __launch_bounds__(256) void rmsnorm_kernel(
    const float* __restrict__ x, const float* __restrict__ w,
    float* __restrict__ y, int Hc)
{
    const int row = blockIdx.x;
    const float* xr = x + (size_t)row * Hc;
    __shared__ float red[256];
    float ss = 0.f;
    for (int j = threadIdx.x; j < Hc; j += 256) { float v = xr[j]; ss += v * v; }
    red[threadIdx.x] = ss;
    __syncthreads();
    for (int s = 128; s > 0; s >>= 1) {
        if (threadIdx.x < s) red[threadIdx.x] += red[threadIdx.x + s];
        __syncthreads();
    }
    float inv = rsqrtf(red[0] / (float)Hc + EPS);
    for (int j = threadIdx.x; j < Hc; j += 256)
        y[(size_t)row * Hc + j] = w[j] * xr[j] * inv;
}

// ---------------------------------------------------------------------------
// Per-head RMSNorm + RoPE + layout split of qkv.  128 threads = 4 waves,
// one 128-dim vector per wave.  24 vectors per token (16 q, 4 k, 4 v).
// ---------------------------------------------------------------------------
__global__ __launch_bounds__(128) void qkrope_kernel(
    const float* __restrict__ qkv,
    const float* __restrict__ cosb, const float* __restrict__ sinb,
    const float* __restrict__ qw, const float* __restrict__ kw,
    float* __restrict__ q, float* __restrict__ kx, float* __restrict__ vx)
{
    const int vid  = blockIdx.x * 4 + (threadIdx.x >> 5);
    const int lane = threadIdx.x & 31;
    const int t = vid / 24, c = vid % 24;
    const float* src = qkv + (size_t)t * QKVD + c * HD;

    float v0[4];
    float ss = 0.f;
#pragma unroll
    for (int j = 0; j < 4; j++) { v0[j] = src[lane * 4 + j]; ss += v0[j] * v0[j]; }
#pragma unroll
    for (int o = 16; o > 0; o >>= 1) ss += __shfl_xor(ss, o, 32);

    if (c < 20) { // q and k: RMSNorm then RoPE on dims [0,64)
        float inv = rsqrtf(ss / (float)HD + EPS);
        const float* w = (c < 16) ? qw : kw;
#pragma unroll
        for (int j = 0; j < 4; j++) v0[j] = w[lane * 4 + j] * v0[j] * inv;
        float r[4];
#pragma unroll
        for (int j = 0; j < 4; j++) r[j] = __shfl_xor(v0[j], 8, 32);
        if (lane < 16) {
#pragma unroll
            for (int j = 0; j < 4; j++) {
                int d = lane * 4 + j;
                float sgn = (d < 32) ? -1.f : 1.f;
                v0[j] = v0[j] * cosb[t * ROT + d] + sgn * r[j] * sinb[t * ROT + d];
            }
        }
    }
    float* dst = (c < 16) ? (q  + ((size_t)c        * Tn + t) * HD)
               : (c < 20) ? (kx + ((size_t)(c - 16) * Tn + t) * HD)
                          : (vx + ((size_t)(c - 20) * Tn + t) * HD);
#pragma unroll
    for (int j = 0; j < 4; j++) dst[lane * 4 + j] = v0[j];
}

// ---------------------------------------------------------------------------
// Attention (no mask).  One block per (head, 16-row q block).
// scores 16x1024 held in LDS; q.kT and P.V via WMMA f32.
// ---------------------------------------------------------------------------
__global__ __launch_bounds__(256) void attn_kernel(
    const float* __restrict__ q, const float* __restrict__ kx,
    const float* __restrict__ vx, float* __restrict__ o)
{
    const int qb = blockIdx.x;           // 64 q blocks
    const int h  = blockIdx.y;           // 16 heads
    const int kvh = h / (NH / NKV);
    __shared__ float qs[16][HD];         // 8 KB
    __shared__ float sc[16][Tn];         // 64 KB
    __shared__ float red[16][16];
    const int tid  = threadIdx.x;
    const int wave = tid >> 5, lane = tid & 31;
    const int half = lane >> 4, idx = lane & 15;
    const float scale = 0.08838834764831845f;   // 1/sqrt(128)

    for (int i = tid; i < 16 * HD; i += 256)
        qs[i >> 7][i & 127] =
            q[((size_t)h * Tn + qb * 16 + (i >> 7)) * HD + (i & 127)] * scale;
    __syncthreads();

    // ---- scores = (q*scale) . k^T ----
    const float* kbase0 = kx + (size_t)kvh * Tn * HD;
    for (int kb = wave; kb < Tn / 16; kb += 8) {
        v8f acc = {};
        const float* kbase = kbase0 + (size_t)kb * 16 * HD;
#pragma unroll
        for (int kk = 0; kk < HD; kk += 4) {
            v2f a, b;
            a[0] = qs[idx][kk + 2 * half];
            a[1] = qs[idx][kk + 2 * half + 1];
            b[0] = kbase[(size_t)idx * HD + kk + 2 * half];
            b[1] = kbase[(size_t)idx * HD + kk + 2 * half + 1];
            acc = __builtin_amdgcn_wmma_f32_16x16x4_f32(
                false, a, false, b, (short)0, acc, false, false);
        }
#pragma unroll
        for (int r = 0; r < 8; r++)
            sc[r + 8 * half][kb * 16 + idx] = acc[r];
    }
    __syncthreads();

    // ---- softmax over 1024 keys, 16 threads per row ----
    {
        const int row = tid >> 4, t16 = tid & 15;
        float mx = -3.4e38f;
        for (int j = t16; j < Tn; j += 16) mx = fmaxf(mx, sc[row][j]);
        red[row][t16] = mx;
        __syncthreads();
        mx = red[row][0];
        for (int i = 1; i < 16; i++) mx = fmaxf(mx, red[row][i]);
        float sum = 0.f;
        for (int j = t16; j < Tn; j += 16) {
            float e = __expf(sc[row][j] - mx);
            sc[row][j] = e; sum += e;
        }
        __syncthreads();
        red[row][t16] = sum;
        __syncthreads();
        float tot = 0.f;
        for (int i = 0; i < 16; i++) tot += red[row][i];
        float inv = 1.f / tot;
        for (int j = t16; j < Tn; j += 16) sc[row][j] *= inv;
    }
    __syncthreads();

    // ---- o = P . V ;  wave owns HD-tile nt = wave (8 tiles of 16) ----
    v8f acc = {};
    const float* vbase = vx + (size_t)kvh * Tn * HD;
    for (int kb = 0; kb < Tn / 16; kb++) {
#pragma unroll
        for (int kk = 0; kk < 16; kk += 4) {
            int krow = kb * 16 + kk + 2 * half;
            v2f a, b;
            a[0] = sc[idx][krow];
            a[1] = sc[idx][krow + 1];
            b[0] = vbase[(size_t)krow * HD + wave * 16 + idx];
            b[1] = vbase[(size_t)(krow + 1) * HD + wave * 16 + idx];
            acc = __builtin_amdgcn_wmma_f32_16x16x4_f32(
                false, a, false, b, (short)0, acc, false, false);
        }
    }
#pragma unroll
    for (int r = 0; r < 8; r++) {
        int m = r + 8 * half;
        int n = wave * 16 + idx;
        o[(size_t)(qb * 16 + m) * (NH * HD) + h * HD + n] = acc[r];
    }
}

// ---------------------------------------------------------------------------
// Router: sigmoid gate, group top-1 (of top-2 sums), then top-4 in group.
// One block per token.
// ---------------------------------------------------------------------------
__global__ __launch_bounds__(256) void router_kernel(
    const float* __restrict__ hbuf, const float* __restrict__ gw,
    const float* __restrict__ bias,
    int* __restrict__ topk_idx, float* __restrict__ topk_w, int* __restrict__ cnt)
{
    const int t = blockIdx.x;
    __shared__ float red[16][17];
    __shared__ float sg[16];
    const int e = threadIdx.x >> 4, j = threadIdx.x & 15;
    const float* hr = hbuf + (size_t)t * Hn;
    const float* gr = gw + (size_t)e * Hn;
    float s = 0.f;
    for (int k = j; k < Hn; k += 16) s += hr[k] * gr[k];
    red[e][j] = s;
    __syncthreads();
    if (j == 0) {
        float tot = 0.f;
        for (int i = 0; i < 16; i++) tot += red[e][i];
        sg[e] = 1.f / (1.f + __expf(-tot));
    }
    __syncthreads();
    if (threadIdx.x == 0) {
        float sfr[En];
        for (int i = 0; i < En; i++) sfr[i] = sg[i] + bias[i];
        float gs[2];
        for (int g = 0; g < 2; g++) {
            float m1 = -1e30f, m2 = -1e30f;
            for (int i = 0; i < 8; i++) {
                float v = sfr[g * 8 + i];
                if (v > m1) { m2 = m1; m1 = v; } else if (v > m2) m2 = v;
            }
            gs[g] = m1 + m2;
        }
        int gsel = (gs[1] > gs[0]) ? 1 : 0;
        float masked[En];
        for (int i = 0; i < En; i++)
            masked[i] = ((i >> 3) == gsel) ? sfr[i] : -1e30f;
        int idx4[Kt];
        for (int kk = 0; kk < Kt; kk++) {
            int bi = 0; float bv = -1e31f;
            for (int i = 0; i < En; i++)
                if (masked[i] > bv) { bv = masked[i]; bi = i; }
            idx4[kk] = bi; masked[bi] = -1e32f;
        }
        float ssum = 0.f;
        for (int kk = 0; kk < Kt; kk++) ssum += sg[idx4[kk]];
        float invs = 1.f / (ssum + 1e-20f);
        for (int kk = 0; kk < Kt; kk++) {
            topk_idx[t * Kt + kk] = idx4[kk];
            topk_w[t * Kt + kk] = sg[idx4[kk]] * invs;   // ROUTED_SCALE = 1
            atomicAdd(&cnt[idx4[kk]], 1);
        }
    }
}

__global__ void zero_int_kernel(int* p, int n) {
    int i = blockIdx.x * blockDim.x + threadIdx.x;
    if (i < n) p[i] = 0;
}

__global__ void scan_kernel(const int* __restrict__ cnt, int* __restrict__ base) {
    if (threadIdx.x == 0) {
        int run = 0;
        for (int e = 0; e < En; e++) { base[e] = run; run += cnt[e]; }
    }
}

__global__ __launch_bounds__(256) void scatter_kernel(
    const int* __restrict__ topk_idx, const float* __restrict__ topk_w,
    const int* __restrict__ base, int* __restrict__ cnt2,
    int* __restrict__ pair_tok, float* __restrict__ pair_w,
    int* __restrict__ slot_for)
{
    int i = blockIdx.x * 256 + threadIdx.x;
    if (i >= NPAIR) return;
    int e = topk_idx[i];
    int pos = atomicAdd(&cnt2[e], 1);
    int slot = base[e] + pos;
    pair_tok[slot] = i >> 2;
    pair_w[slot]   = topk_w[i];
    slot_for[i]    = slot;
}

// ---------------------------------------------------------------------------
// MoE gate/up GEMM with row gather:  Cout[slot, :MIn] = h[tok(slot)] . W_e^T
// blockIdx.y = e*64 + tile.  Inactive tiles exit early.  Async-LDS staged.
// Padded rows compute garbage on valid memory and are store-masked.
// ---------------------------------------------------------------------------
__global__ __launch_bounds__(256) void moe_gemm_gather(
    const float* __restrict__ hbuf, const float* __restrict__ W,
    float* __restrict__ Cout,
    const int* __restrict__ pair_tok,
    const int* __restrict__ base, const int* __restrict__ cnt)
{
    const int e = blockIdx.y >> 6, b = blockIdx.y & 63;
    int nrows = cnt[e] - b * 16;
    if (nrows <= 0) return;
    if (nrows > 16) nrows = 16;
    const int start = base[e] + b * 16;
    const int n0 = blockIdx.x * 128;
    const float* Bm = W + (size_t)e * MIn * Hn;

    __shared__ __align__(16) float As[2][16][36];
    __shared__ __align__(16) float Bs[2][128][36];
    __shared__ int toks[16];
    const int tid = threadIdx.x;
    const int wave = tid >> 5, lane = tid & 31;
    const int half = lane >> 4, idx = lane & 15;
    if (tid < 16) toks[tid] = (tid < nrows) ? pair_tok[start + tid] : pair_tok[start];
    __syncthreads();

    const int ar = (tid << 1) >> 5, ac = (tid << 1) & 31;
    const float* arow = hbuf + (size_t)toks[ar] * Hn + ac;

    auto issue = [&](int buf, int k0) {
        async_copy_b64(arow + k0, &As[buf][ar][ac]);
#pragma unroll
        for (int i = 0; i < 4; i++) {
            int f = tid + i * 256;
            int r = f >> 3, c4 = (f & 7) << 2;
            async_copy_b128(Bm + (size_t)(n0 + r) * Hn + k0 + c4, &Bs[buf][r][c4]);
        }
    };

    v8f acc = {};
    const int nch = Hn >> 5;
    issue(0, 0);
    for (int ch = 0; ch < nch; ch++) {
        wait_async0();
        __syncthreads();
        if (ch + 1 < nch) issue((ch + 1) & 1, (ch + 1) << 5);
        const int buf = ch & 1;
#pragma unroll
        for (int kk = 0; kk < 32; kk += 4) {
            v2f a, bb;
            a[0] = As[buf][idx][kk + 2 * half];
            a[1] = As[buf][idx][kk + 2 * half + 1];
            bb[0] = Bs[buf][wave * 16 + idx][kk + 2 * half];
            bb[1] = Bs[buf][wave * 16 + idx][kk + 2 * half + 1];
            acc = __builtin_amdgcn_wmma_f32_16x16x4_f32(
                false, a, false, bb, (short)0, acc, false, false);
        }
    }
#pragma unroll
    for (int r = 0; r < 8; r++) {
        int row = r + 8 * half;
        if (row < nrows)
            Cout[(size_t)(start + row) * MIn + n0 + wave * 16 + idx] = acc[r];
    }
}

__global__ __launch_bounds__(256) void silu_mul_kernel(
    const float* __restrict__ g, const float* __restrict__ u,
    float* __restrict__ a, int n)
{
    int i = blockIdx.x * 256 + threadIdx.x;
    if (i < n) {
        float x = g[i];
        a[i] = (x / (1.f + __expf(-x))) * u[i];
    }
}

// ---------------------------------------------------------------------------
// MoE down projection: pairout[slot, :Hn] = acts[slot] . ed_e^T
// ---------------------------------------------------------------------------
__global__ __launch_bounds__(256) void moe_down_kernel(
    const float* __restrict__ acts, const float* __restrict__ ed,
    float* __restrict__ pairout,
    const int* __restrict__ base, const int* __restrict__ cnt)
{
    const int e = blockIdx.y >> 6, b = blockIdx.y & 63;
    int nrows = cnt[e] - b * 16;
    if (nrows <= 0) return;
    if (nrows > 16) nrows = 16;
    const int start = base[e] + b * 16;
    const int n0 = blockIdx.x * 128;
    const float* Bm = ed + (size_t)e * Hn * MIn;

    __shared__ __align__(16) float As[2][16][36];
    __shared__ __align__(16) float Bs[2][128][36];
    const int tid = threadIdx.x;
    const int wave = tid >> 5, lane = tid & 31;
    const int half = lane >> 4, idx = lane & 15;

    const int ar = (tid << 1) >> 5, ac = (tid << 1) & 31;
    int arowi = start + ar; if (arowi > NPAIR - 1) arowi = NPAIR - 1;
    const float* arow = acts + (size_t)arowi * MIn + ac;

    auto issue = [&](int buf, int k0) {
        async_copy_b64(arow + k0, &As[buf][ar][ac]);
#pragma unroll
        for (int i = 0; i < 4; i++) {
            int f = tid + i * 256;
            int r = f >> 3, c4 = (f & 7) << 2;
            async_copy_b128(Bm + (size_t)(n0 + r) * MIn + k0 + c4, &Bs[buf][r][c4]);
        }
    };

    v8f acc = {};
    const int nch = MIn >> 5;
    issue(0, 0);
    for (int ch = 0; ch < nch; ch++) {
        wait_async0();
        __syncthreads();
        if (ch + 1 < nch) issue((ch + 1) & 1, (ch + 1) << 5);
        const int buf = ch & 1;
#pragma unroll
        for (int kk = 0; kk < 32; kk += 4) {
            v2f a, bb;
            a[0] = As[buf][idx][kk + 2 * half];
            a[1] = As[buf][idx][kk + 2 * half + 1];
            bb[0] = Bs[buf][wave * 16 + idx][kk + 2 * half];
            bb[1] = Bs[buf][wave * 16 + idx][kk + 2 * half + 1];
            acc = __builtin_amdgcn_wmma_f32_16x16x4_f32(
                false, a, false, bb, (short)0, acc, false, false);
        }
    }
#pragma unroll
    for (int r = 0; r < 8; r++) {
        int row = r + 8 * half;
        if (row < nrows)
            pairout[(size_t)(start + row) * Hn + n0 + wave * 16 + idx] = acc[r];
    }
}

// Deterministic per-token combine of the 4 routed-expert outputs.
__global__ __launch_bounds__(256) void combine_kernel(
    const float* __restrict__ pairout, const int* __restrict__ slot_for,
    const float* __restrict__ topk_w, float* __restrict__ routed)
{
    int i = blockIdx.x * 256 + threadIdx.x;      // T*Hn elements
    int t = i >> 11, col = i & (Hn - 1);
    float s = 0.f;
#pragma unroll
    for (int j = 0; j < Kt; j++) {
        int slot = slot_for[t * Kt + j];
        s += topk_w[t * Kt + j] * pairout[(size_t)slot * Hn + col];
    }
    routed[i] = s;
}

// ---------------------------------------------------------------------------
// Launch
// ---------------------------------------------------------------------------
extern "C" void kernel_launch(void* const* d_in, const int* in_sizes, int n_in,
                              void* d_out, int out_size, void* d_ws, size_t ws_size,
                              hipStream_t stream)
{
    const float* hidden = (const float*)d_in[0];
    const float* cosb   = (const float*)d_in[1];
    const float* sinb   = (const float*)d_in[2];
    const float* ln1w   = (const float*)d_in[3];
    const float* w_qkv  = (const float*)d_in[4];
    const float* qlnw   = (const float*)d_in[5];
    const float* klnw   = (const float*)d_in[6];
    const float* w_dense= (const float*)d_in[7];
    const float* ln2w   = (const float*)d_in[8];
    const float* gate_w = (const float*)d_in[9];
    const float* ebias  = (const float*)d_in[10];
    const float* eg     = (const float*)d_in[11];
    const float* eu     = (const float*)d_in[12];
    const float* ed     = (const float*)d_in[13];
    const float* sg     = (const float*)d_in[14];
    const float* su     = (const float*)d_in[15];
    const float* sd     = (const float*)d_in[16];
    float* out = (float*)d_out;

    float* ws = (float*)d_ws;
    float* xn     = ws; ws += (size_t)Tn * Hn;
    float* qkvb   = ws; ws += (size_t)Tn * QKVD;
    float* qbuf   = ws; ws += (size_t)NH * Tn * HD;
    float* kbuf   = ws; ws += (size_t)NKV * Tn * HD;
    float* vbuf   = ws; ws += (size_t)NKV * Tn * HD;
    float* aout   = ws; ws += (size_t)Tn * Hn;
    float* x1     = ws; ws += (size_t)Tn * Hn;
    float* h2     = ws; ws += (size_t)Tn * Hn;
    float* routed = ws; ws += (size_t)Tn * Hn;
    float* gbuf   = ws; ws += (size_t)NPAIR * MIn;
    float* ubuf   = ws; ws += (size_t)NPAIR * MIn;
    float* actb   = ws; ws += (size_t)NPAIR * MIn;
    float* pouts  = ws; ws += (size_t)NPAIR * Hn;
    float* gsb    = ws; ws += (size_t)Tn * MIn;
    float* usb    = ws; ws += (size_t)Tn * MIn;
    float* asb    = ws; ws += (size_t)Tn * MIn;
    float* topkw  = ws; ws += NPAIR;
    float* pairw  = ws; ws += NPAIR;
    int* ip = (int*)ws;
    int* topki   = ip; ip += NPAIR;
    int* pairtok = ip; ip += NPAIR;
    int* slotfor = ip; ip += NPAIR;
    int* cnt     = ip; ip += En;
    int* cnt2    = ip; ip += En;
    int* baseb   = ip; ip += En;

    // ---- attention branch ----
    rmsnorm_kernel<<<Tn, 256, 0, stream>>>(hidden, ln1w, xn, Hn);
    gemm16_kernel<GM_STORE><<<dim3(QKVD / 128, Tn / 16), 256, 0, stream>>>(
        xn, w_qkv, qkvb, Tn, QKVD, Hn, nullptr, nullptr);
    qkrope_kernel<<<(Tn * 24) / 4, 128, 0, stream>>>(
        qkvb, cosb, sinb, qlnw, klnw, qbuf, kbuf, vbuf);
    attn_kernel<<<dim3(Tn / 16, NH), 256, 0, stream>>>(qbuf, kbuf, vbuf, aout);
    gemm16_kernel<GM_ADD><<<dim3(Hn / 128, Tn / 16), 256, 0, stream>>>(
        aout, w_dense, x1, Tn, Hn, Hn, hidden, nullptr);

    // ---- MoE branch ----
    rmsnorm_kernel<<<Tn, 256, 0, stream>>>(x1, ln2w, h2, Hn);
    zero_int_kernel<<<1, 64, 0, stream>>>(cnt, 2 * En);   // cnt and cnt2
    router_kernel<<<Tn, 256, 0, stream>>>(h2, gate_w, ebias, topki, topkw, cnt);
    scan_kernel<<<1, 1, 0, stream>>>(cnt, baseb);
    scatter_kernel<<<NPAIR / 256, 256, 0, stream>>>(
        topki, topkw, baseb, cnt2, pairtok, pairw, slotfor);

    moe_gemm_gather<<<dim3(MIn / 128, En * 64), 256, 0, stream>>>(
        h2, eg, gbuf, pairtok, baseb, cnt);
    moe_gemm_gather<<<dim3(MIn / 128, En * 64), 256, 0, stream>>>(
        h2, eu, ubuf, pairtok, baseb, cnt);
    silu_mul_kernel<<<(NPAIR * MIn) / 256, 256, 0, stream>>>(
        gbuf, ubuf, actb, NPAIR * MIn);
    moe_down_kernel<<<dim3(Hn / 128, En * 64), 256, 0, stream>>>(
        actb, ed, pouts, baseb, cnt);
    combine_kernel<<<(Tn * Hn) / 256, 256, 0, stream>>>(
        pouts, slotfor, topkw, routed);

    // ---- shared expert + final residual fusion ----
    gemm16_kernel<GM_STORE><<<dim3(MIn / 128, Tn / 16), 256, 0, stream>>>(
        h2, sg, gsb, Tn, MIn, Hn, nullptr, nullptr);
    gemm16_kernel<GM_STORE><<<dim3(MIn / 128, Tn / 16), 256, 0, stream>>>(
        h2, su, usb, Tn, MIn, Hn, nullptr, nullptr);
    silu_mul_kernel<<<(Tn * MIn) / 256, 256, 0, stream>>>(gsb, usb, asb, Tn * MIn);
    gemm16_kernel<GM_FINAL><<<dim3(Hn / 128, Tn / 16), 256, 0, stream>>>(
        asb, sd, out, Tn, Hn, MIn, x1, routed);
}